// Model_ViewPoint_77506979823876
// MI455X (gfx1250) — compile-verified
//
#include <hip/hip_runtime.h>

typedef __attribute__((ext_vector_type(2))) float v2f;
typedef __attribute__((ext_vector_type(8))) float v8f;

// Wave-level sum using V_WMMA_F32_16X16X4_F32.
// A (16x4 f32) = 2 VGPRs/lane: VGPR0 = per-lane partial, VGPR1 = 0.
// B (4x16 f32) = all ones. D[m][n] = rowsum_m for every n, so
// (sum of the 8 C/D VGPRs in lane k) + (same in lane k^16) == wave total.
__device__ __forceinline__ float wave_sum_wmma(float v) {
    v2f a; a[0] = v;   a[1] = 0.0f;
    v2f b; b[0] = 1.0f; b[1] = 1.0f;
    v8f c = {};
    c = __builtin_amdgcn_wmma_f32_16x16x4_f32(
        /*neg_a=*/false, a, /*neg_b=*/false, b,
        /*c_mod=*/(short)0, c, /*reuse_a=*/false, /*reuse_b=*/false);
    float s = ((c[0] + c[1]) + (c[2] + c[3])) + ((c[4] + c[5]) + (c[6] + c[7]));
    s += __shfl_xor(s, 16, 32);   // combine rows 0..7 with rows 8..15
    return s;                     // every lane holds the wave total
}

__device__ __forceinline__ float pt_term(float X, float Y, float RX, float RY, float W) {
    float x = (X + 1.0f) * 0.5f;
    float y = 1.0f - (Y + 1.0f) * 0.5f;
    float dx = x - RX;
    float dy = y - RY;
    return W * sqrtf(dx * dx + dy * dy);
}

// Stage 1: each thread consumes 4 consecutive points via pure float4 (b128) loads,
// wave-reduces with WMMA, block-reduces through LDS, writes one partial per block.
__global__ void vp_loss_partial(const float* __restrict__ kp3d,
                                const float* __restrict__ ref,
                                const float* __restrict__ wgt,
                                float* __restrict__ partials,
                                int nquads, int npts) {
    const float4* kp4 = reinterpret_cast<const float4*>(kp3d);
    const float4* rf4 = reinterpret_cast<const float4*>(ref);
    const float4* w4  = reinterpret_cast<const float4*>(wgt);

    float acc = 0.0f;
    const int gtid   = blockIdx.x * blockDim.x + threadIdx.x;
    const int stride = gridDim.x * blockDim.x;

    for (int q = gtid; q < nquads; q += stride) {
        // 4 points = 12 kp floats, 8 ref floats, 4 weights -> six b128 loads
        float4 k0 = kp4[3 * q + 0];   // x0 y0 z0 x1
        float4 k1 = kp4[3 * q + 1];   // y1 z1 x2 y2
        float4 k2 = kp4[3 * q + 2];   // z2 x3 y3 z3
        float4 r0 = rf4[2 * q + 0];   // rx0 ry0 rx1 ry1
        float4 r1 = rf4[2 * q + 1];   // rx2 ry2 rx3 ry3
        float4 wt = w4[q];

        acc += pt_term(k0.x, k0.y, r0.x, r0.y, wt.x);
        acc += pt_term(k0.w, k1.x, r0.z, r0.w, wt.y);
        acc += pt_term(k1.z, k1.w, r1.x, r1.y, wt.z);
        acc += pt_term(k2.y, k2.z, r1.z, r1.w, wt.w);
    }

    // Tail points (npts not divisible by 4) — not hit for 1024x4096 but kept general.
    int ti = nquads * 4 + gtid;
    if (ti < npts) {
        acc += pt_term(kp3d[3 * ti + 0], kp3d[3 * ti + 1],
                       ref[2 * ti + 0], ref[2 * ti + 1], wgt[ti]);
    }

    float ws = wave_sum_wmma(acc);   // EXEC all-ones here: no divergence at this point

    __shared__ float lds[8];
    const int lane = threadIdx.x & 31;
    const int wave = threadIdx.x >> 5;
    if (lane == 0) lds[wave] = ws;
    __syncthreads();
    if (threadIdx.x == 0) {
        float t = 0.0f;
        const int nw = blockDim.x >> 5;
        for (int i = 0; i < nw; ++i) t += lds[i];
        partials[blockIdx.x] = t;
    }
}

// Stage 2: one block folds all block partials into the scalar output.
// Fixed traversal + WMMA/LDS tree => bitwise-deterministic across replays.
__global__ void vp_loss_final(const float* __restrict__ partials, int n,
                              float* __restrict__ out) {
    float acc = 0.0f;
    for (int i = threadIdx.x; i < n; i += blockDim.x) acc += partials[i];

    float ws = wave_sum_wmma(acc);

    __shared__ float lds[8];
    const int lane = threadIdx.x & 31;
    const int wave = threadIdx.x >> 5;
    if (lane == 0) lds[wave] = ws;
    __syncthreads();
    if (threadIdx.x == 0) {
        float t = 0.0f;
        const int nw = blockDim.x >> 5;
        for (int i = 0; i < nw; ++i) t += lds[i];
        out[0] = t;
    }
}

extern "C" void kernel_launch(void* const* d_in, const int* in_sizes, int n_in,
                              void* d_out, int out_size, void* d_ws, size_t ws_size,
                              hipStream_t stream) {
    const float* kp3d = (const float*)d_in[0];   // (B,N,3) f32
    const float* ref  = (const float*)d_in[1];   // (B,N,2) f32
    const float* wgt  = (const float*)d_in[2];   // (B,N)   f32

    const int npts   = in_sizes[2];              // B*N = 4,194,304
    const int nquads = npts / 4;

    float* partials = (float*)d_ws;              // needs blocks*4 bytes (16 KiB)

    const int threads = 256;                     // 8 wave32 per block
    int blocks = (nquads + threads - 1) / threads;
    if (blocks < 1) blocks = 1;
    if (blocks > 4096) blocks = 4096;            // cap d_ws footprint; grid-stride covers rest

    vp_loss_partial<<<blocks, threads, 0, stream>>>(kp3d, ref, wgt, partials, nquads, npts);
    vp_loss_final<<<1, 256, 0, stream>>>(partials, blocks, (float*)d_out);
}